// HashEmbedders_17179869184521
// MI455X (gfx1250) — compile-verified
//
#include <hip/hip_runtime.h>

// ---------------------------------------------------------------------------
// Multi-resolution hash-grid encoding (Instant-NGP style), MI455X / gfx1250.
// Memory-gather bound: ~1 GB of random 8-byte gathers over 64 MB of tables
// that stay resident in the 192 MB L2; ~204 MB of HBM traffic total.
// No matmul structure -> WMMA would only add data movement. gfx1250 paths
// used instead: async global->LDS coordinate staging (ASYNCcnt +
// s_wait_asynccnt), 8 in-flight b64 gathers/thread, NT b64 output stores.
// ---------------------------------------------------------------------------

#define NLV        16
#define TMASK      ((1u << 19) - 1u)
#define TBL_FLOATS (1u << 20)        // 2^19 entries * 2 floats per level
#define PTS_PER_BLK 16
#define BLK        256

typedef float __attribute__((ext_vector_type(2))) f32x2;

#if defined(__gfx1250__) && \
    __has_builtin(__builtin_amdgcn_global_load_async_to_lds_b32) && \
    __has_builtin(__builtin_amdgcn_s_wait_asynccnt)
#define USE_ASYNC_LDS 1
#else
#define USE_ASYNC_LDS 0
#endif

// res_i = floor(16 * 2^(i/3))  (b = exp(ln(512/16)/15) = 2^(1/3) exactly;
// float32(b) rounds slightly above 2^(1/3), so power-of-two levels floor to
// exactly 32/64/.../512 in the reference too). Varying per-lane index ->
// keep as a 64B memory table (L0-resident).
__constant__ float c_res[NLV] = {
    16.f, 20.f, 25.f, 32.f, 40.f, 50.f, 64.f, 80.f,
    101.f, 128.f, 161.f, 203.f, 256.f, 322.f, 406.f, 512.f
};

__global__ __launch_bounds__(BLK, 8)
void hashgrid_fwd_kernel(const float* __restrict__ x,
                         const float* __restrict__ tables,
                         f32x2* __restrict__ out,
                         int n_points)
{
    // spatial-hash primes: compile-time immediates (not __constant__, so they
    // fold into the v_mul operands instead of being loaded).
    constexpr unsigned PR0 = 1u, PR1 = 2654435761u, PR2 = 805459861u;

    __shared__ float s_x[PTS_PER_BLK * 3];   // 16 points * 3 coords

    const int tid = threadIdx.x;
    const int pt0 = blockIdx.x * PTS_PER_BLK;

    // ---- stage 48 coordinate floats into LDS (coalesced, once per block) ----
    if (tid < PTS_PER_BLK * 3) {
        const int g = pt0 * 3 + tid;
        if (g < n_points * 3) {
#if USE_ASYNC_LDS
            __builtin_amdgcn_global_load_async_to_lds_b32(
                (__attribute__((address_space(1))) int*)(x + g),
                (__attribute__((address_space(3))) int*)(&s_x[tid]),
                /*imm offset*/0, /*cpol*/0);
#else
            s_x[tid] = x[g];
#endif
        }
    }
#if USE_ASYNC_LDS
    __builtin_amdgcn_s_wait_asynccnt(0);
#endif
    __syncthreads();

    // thread = (point, level); level fastest => wave32 = 2 points x 16 levels,
    // so the final f32x2 stores are 32 consecutive 8B slots (256B/wave).
    const int level = tid & (NLV - 1);
    const int lp    = tid >> 4;
    const int n     = pt0 + lp;
    if (n >= n_points) return;

    const float res   = c_res[level];
    const float gridw = 2.0f / res;          // (box_max-box_min)/res, f32 div

    float    w[3];
    unsigned hv[3][2];
#pragma unroll
    for (int d = 0; d < 3; ++d) {
        float xc = s_x[lp * 3 + d];
        xc = fminf(fmaxf(xc, -1.0f), 1.0f);
        const float bl   = floorf((xc + 1.0f) / gridw);   // matches reference fp ops
        const float vmin = bl * gridw - 1.0f;
        const float vmax = vmin + gridw;
        w[d] = (xc - vmin) / (vmax - vmin);
        const unsigned u = (unsigned)bl;                  // bl >= 0 after clamp
        const unsigned p = (d == 0) ? PR0 : (d == 1) ? PR1 : PR2;
        hv[d][0] = u * p;
        hv[d][1] = (u + 1u) * p;
    }

    // ---- 8 corner hashes + faithful w**offset weights ----
    unsigned idx[8];
    float    wc[8];
#pragma unroll
    for (int k = 0; k < 8; ++k) {
        const int o0 = (k >> 2) & 1;   // dim0 slowest (itertools.product order)
        const int o1 = (k >> 1) & 1;
        const int o2 =  k       & 1;
        idx[k] = (hv[0][o0] ^ hv[1][o1] ^ hv[2][o2]) & TMASK;
        float p = o0 ? w[0] : 1.0f;
        p      *= o1 ? w[1] : 1.0f;
        p      *= o2 ? w[2] : 1.0f;
        wc[k] = p;                     // corner 0 -> weight 1.0 (faithful)
    }

    // ---- 8 independent b64 gathers (all in flight before any use) ----
    const f32x2* __restrict__ tbl =
        (const f32x2*)(tables + (unsigned)level * TBL_FLOATS);
    f32x2 e[8];
#pragma unroll
    for (int k = 0; k < 8; ++k) e[k] = tbl[idx[k]];

    float f0 = 0.0f, f1 = 0.0f;
#pragma unroll
    for (int k = 0; k < 8; ++k) {
        f0 += wc[k] * e[k].x;
        f1 += wc[k] * e[k].y;
    }

    // NT store: output is 128 MB (> L2 share); keep hash tables L2-resident.
    f32x2 r;
    r.x = f0;
    r.y = f1;
    __builtin_nontemporal_store(r, &out[(size_t)n * NLV + level]);
}

extern "C" void kernel_launch(void* const* d_in, const int* in_sizes, int n_in,
                              void* d_out, int out_size, void* d_ws, size_t ws_size,
                              hipStream_t stream)
{
    const float* x      = (const float*)d_in[0];   // [N,3] f32
    const float* tables = (const float*)d_in[1];   // [16, 2^19, 2] f32
    f32x2*       out    = (f32x2*)d_out;           // [N, 32] f32 = [N,16] f32x2

    const int n_points = in_sizes[0] / 3;
    const int blocks   = (n_points + PTS_PER_BLK - 1) / PTS_PER_BLK;

    hashgrid_fwd_kernel<<<blocks, BLK, 0, stream>>>(x, tables, out, n_points);
}